// EquivEncoder_41652592837139
// MI455X (gfx1250) — compile-verified
//
#include <hip/hip_runtime.h>
#include <hip/hip_bf16.h>

// ---------------- problem constants (match reference) ----------------
#define NN      10000            // nodes
#define NE      160000           // edges
#define NG      64               // graphs
#define DIM     128              // hidden dim
#define NL      4                // layers
#define HIN     100              // input feature dim
#define NRBF    64               // D/2
#define D2      (2*DIM)          // 256
#define D3      (3*DIM)          // 384
#define ECHUNK  32000            // edge chunk for W GEMM (multiple of 16, divides NE)
#define PI_F    3.14159265358979323846f
#define CUT     5.0f

#define KC      64               // K chunk staged in LDS
#define LDA_S   68               // padded LDS row (floats) for A panel
#define LDB_S   80               // padded LDS row (floats) for B panel

typedef __attribute__((ext_vector_type(2))) float v2f;
typedef __attribute__((ext_vector_type(8))) float v8f;

__device__ __forceinline__ float sigmoid_f(float x) { return 1.0f / (1.0f + expf(-x)); }
__device__ __forceinline__ float silu_f(float x)    { return x * sigmoid_f(x); }

// =====================================================================
// Block-cooperative fp32 WMMA GEMM with async global->LDS staging.
//   C[M,Nc] = A[M,K] * B[K,Nc] (+bias) (+silu)
// Row-major, lda=K, ldb=ldc=Nc. Requires K%4==0, M%16==0, Nc%64==0.
// Block = 256 threads (8 waves) computes a 128x64 tile:
//   wave w owns rows [m0+16w, m0+16w+16), all waves share the B strip.
// K is staged in 64-wide chunks via GLOBAL_LOAD_ASYNC_TO_LDS_B128
// (ASYNCcnt), fragments consumed from LDS via ds loads.
// Fragment maps (V_WMMA_F32_16X16X4_F32, ISA 7.12.2):
//   A 16x4 : lane l -> m=l%16, k=(l/16)*2 + reg
//   B 4x16 : lane l -> n=l%16, k=(l/16)*2 + reg
//   C 16x16: reg r, lane l -> m=r+(l/16)*8, n=l%16
// =====================================================================
template <int EPI>   // 0 = none, 1 = silu
__global__ __launch_bounds__(256)
void wmma_gemm_f32(const float* __restrict__ A, const float* __restrict__ B,
                   const float* __restrict__ bias, float* __restrict__ C,
                   int M, int K, int Nc)
{
    __shared__ float As[128 * LDA_S];
    __shared__ float Bs[KC * LDB_S];

    const int lane = threadIdx.x & 31;
    const int wave = threadIdx.x >> 5;
    const int n0 = blockIdx.x << 6;
    const int m0 = blockIdx.y << 7;

    const int la = lane & 15;
    const int kh = (lane >> 4) << 1;        // 0 or 2
    const int wrow = wave << 4;
    const bool valid = (m0 + wrow) < M;     // M%16==0 -> tiles all-or-nothing
    const int Mm1 = M - 1;

    v8f acc[4] = {v8f{}, v8f{}, v8f{}, v8f{}};

    for (int kc = 0; kc < K; kc += KC) {
        const int klen = (K - kc < KC) ? (K - kc) : KC;   // always %4==0
        const int qA = klen >> 2;                          // b128 per A row

        // ---- async stage A panel: 128 rows x klen, 16B granules ----
        for (int i = threadIdx.x; i < 128 * qA; i += 256) {
            int row = i / qA, q = i - row * qA;
            int rowg = m0 + row; if (rowg > Mm1) rowg = Mm1;
            unsigned lds = (unsigned)(uintptr_t)&As[row * LDA_S + 4 * q];
            unsigned long long ga =
                (unsigned long long)(uintptr_t)(A + (size_t)rowg * K + kc + 4 * q);
            asm volatile("global_load_async_to_lds_b128 %0, %1, off"
                         :: "v"(lds), "v"(ga) : "memory");
        }
        // ---- async stage B panel: klen rows x 64 ----
        for (int i = threadIdx.x; i < klen * 16; i += 256) {
            int kr = i >> 4, q = i & 15;
            unsigned lds = (unsigned)(uintptr_t)&Bs[kr * LDB_S + 4 * q];
            unsigned long long ga =
                (unsigned long long)(uintptr_t)(B + (size_t)(kc + kr) * Nc + n0 + 4 * q);
            asm volatile("global_load_async_to_lds_b128 %0, %1, off"
                         :: "v"(lds), "v"(ga) : "memory");
        }
        asm volatile("s_wait_asynccnt 0x0" ::: "memory");
        __syncthreads();

        if (valid) {
            const float* __restrict__ ap0 = &As[(wrow + la) * LDA_S + kh];
            if (klen == KC) {
                // compile-time trip count: let DS loads of later iterations
                // issue underneath the WMMA accumulation chain
#pragma unroll 4
                for (int k = 0; k < KC; k += 4) {
                    v2f a;
                    a.x = ap0[k];
                    a.y = ap0[k + 1];
                    const float* __restrict__ b0 = &Bs[(k + kh) * LDB_S + la];
#pragma unroll
                    for (int j = 0; j < 4; ++j) {
                        v2f b;
                        b.x = b0[16 * j];
                        b.y = b0[LDB_S + 16 * j];
                        acc[j] = __builtin_amdgcn_wmma_f32_16x16x4_f32(
                            false, a, false, b, (short)0, acc[j], false, false);
                    }
                }
            } else {
                for (int k = 0; k < klen; k += 4) {
                    v2f a;
                    a.x = ap0[k];
                    a.y = ap0[k + 1];
                    const float* __restrict__ b0 = &Bs[(k + kh) * LDB_S + la];
#pragma unroll
                    for (int j = 0; j < 4; ++j) {
                        v2f b;
                        b.x = b0[16 * j];
                        b.y = b0[LDB_S + 16 * j];
                        acc[j] = __builtin_amdgcn_wmma_f32_16x16x4_f32(
                            false, a, false, b, (short)0, acc[j], false, false);
                    }
                }
            }
        }
        __syncthreads();
    }

    if (!valid) return;
    const int mrow = m0 + wrow + ((lane >> 4) << 3);
    const int col0 = n0 + la;
#pragma unroll
    for (int j = 0; j < 4; ++j) {
        const int c = col0 + 16 * j;
        const float bv = bias ? bias[c] : 0.0f;
#pragma unroll
        for (int r = 0; r < 8; ++r) {
            float x = acc[j][r] + bv;
            if (EPI == 1) x = silu_f(x);
            C[(size_t)(mrow + r) * Nc + c] = x;
        }
    }
}

// ---------------- Fourier time embedding: t_emb[g, 0:64]=cos, [64:128]=sin
__global__ void temb_kernel(const float* __restrict__ t, const float* __restrict__ fW,
                            float* __restrict__ temb)
{
    int i = blockIdx.x * blockDim.x + threadIdx.x;
    if (i >= NG * DIM) return;
    int g = i / DIM, j = i % DIM;
    float x = 2.0f * PI_F * t[g] * fW[j & (NRBF - 1)];
    temb[i] = (j < NRBF) ? cosf(x) : sinf(x);
}

// ---------------- s += tproj[node_index[n]]  (tproj already has ntp_b)
__global__ void add_tproj_kernel(float* __restrict__ s, const float* __restrict__ tproj,
                                 const int* __restrict__ node_index)
{
    int i = blockIdx.x * blockDim.x + threadIdx.x;
    if (i >= NN * DIM) return;
    int n = i / DIM, d = i % DIM;
    s[i] += tproj[(size_t)node_index[n] * DIM + d];
}

__global__ void zero_kernel(float* __restrict__ p, int n)
{
    int i = blockIdx.x * blockDim.x + threadIdx.x;
    if (i < n) p[i] = 0.0f;
}

// ---------------- edge embedding: unit vectors + Bessel RBF * cosine cutoff
__global__ void edge_embed_kernel(const float* __restrict__ pos, const int* __restrict__ eidx,
                                  float* __restrict__ edge_states, float* __restrict__ unit)
{
    int e = blockIdx.x * blockDim.x + threadIdx.x;
    if (e >= NE) return;
    int src = eidx[e];
    int dst = eidx[NE + e];
    float rx = pos[dst * 3 + 0] - pos[src * 3 + 0];
    float ry = pos[dst * 3 + 1] - pos[src * 3 + 1];
    float rz = pos[dst * 3 + 2] - pos[src * 3 + 2];
    float dist = sqrtf(rx * rx + ry * ry + rz * rz + 1e-12f);
    float inv = 1.0f / dist;
    unit[e * 3 + 0] = rx * inv;
    unit[e * 3 + 1] = ry * inv;
    unit[e * 3 + 2] = rz * inv;
    float fc = (dist < CUT) ? 0.5f * (cosf(PI_F * dist / CUT) + 1.0f) : 0.0f;
    float scale = fc * inv;
    float base = PI_F * dist / CUT;
    float* es = edge_states + (size_t)e * NRBF;
    for (int j = 0; j < NRBF; ++j)
        es[j] = sinf((float)(j + 1) * base) * scale;
}

// ---------------- EquivLayerNorm (wave per node) + seed scatter buffers
__global__ __launch_bounds__(256)
void norm_kernel(const float* __restrict__ s, const float* __restrict__ v,
                 const float* __restrict__ ln_g, const float* __restrict__ ln_b,
                 const float* __restrict__ ln_gv,
                 float* __restrict__ sn, float* __restrict__ vn,
                 float* __restrict__ s2, float* __restrict__ v2)
{
    const int lane = threadIdx.x & 31;
    const int node = blockIdx.x * 8 + (threadIdx.x >> 5);
    if (node >= NN) return;

    float sv[4], sum = 0.0f, sum2 = 0.0f;
#pragma unroll
    for (int i = 0; i < 4; ++i) {
        float x = s[(size_t)node * DIM + lane + 32 * i];
        sv[i] = x; sum += x; sum2 += x * x;
    }
#pragma unroll
    for (int o = 16; o >= 1; o >>= 1) {
        sum  += __shfl_xor(sum, o, 32);
        sum2 += __shfl_xor(sum2, o, 32);
    }
    float mu = sum * (1.0f / DIM);
    float var = sum2 * (1.0f / DIM) - mu * mu;
    float rstd = rsqrtf(var + 1e-5f);
#pragma unroll
    for (int i = 0; i < 4; ++i) {
        int d = lane + 32 * i;
        float x = (sv[i] - mu) * rstd * ln_g[d] + ln_b[d];
        sn[(size_t)node * DIM + d] = x;
        s2[(size_t)node * DIM + d] = x;
    }

    float vv[12], vsum = 0.0f;
#pragma unroll
    for (int c = 0; c < 3; ++c)
#pragma unroll
        for (int i = 0; i < 4; ++i) {
            float x = v[((size_t)node * 3 + c) * DIM + lane + 32 * i];
            vv[c * 4 + i] = x; vsum += x * x;
        }
#pragma unroll
    for (int o = 16; o >= 1; o >>= 1) vsum += __shfl_xor(vsum, o, 32);
    float rr = rsqrtf(vsum * (1.0f / DIM) + 1e-5f);
#pragma unroll
    for (int c = 0; c < 3; ++c)
#pragma unroll
        for (int i = 0; i < 4; ++i) {
            int d = lane + 32 * i;
            float x = vv[c * 4 + i] * rr * ln_gv[d];
            vn[((size_t)node * 3 + c) * DIM + d] = x;
            v2[((size_t)node * 3 + c) * DIM + d] = x;
        }
}

// ---------------- per-edge message + gate + scatter (wave per edge)
__global__ __launch_bounds__(256)
void edge_interact_kernel(const float* __restrict__ Wc, const float* __restrict__ phi,
                          const float* __restrict__ vn, const float* __restrict__ unit,
                          const int* __restrict__ eidx,
                          const float* __restrict__ ei_w, const float* __restrict__ ei_b,
                          float* __restrict__ s2, float* __restrict__ v2,
                          int e_base, int e_count)
{
    const int lane = threadIdx.x & 31;
    const int er = blockIdx.x * 8 + (threadIdx.x >> 5);
    if (er >= e_count) return;
    const int e = e_base + er;
    const int src = eidx[e];
    const int dst = eidx[NE + e];
    const float* __restrict__ We = Wc + (size_t)er * D3;
    const float* __restrict__ ph = phi + (size_t)src * D3;

    float ps[4], pvv[4], pvs[4];
    float dot = 0.0f;
#pragma unroll
    for (int i = 0; i < 4; ++i) {
        int d = lane + 32 * i;
        ps[i]  = We[d]           * ph[d];
        pvv[i] = We[DIM + d]     * ph[DIM + d];
        pvs[i] = We[2 * DIM + d] * ph[2 * DIM + d];
        dot += ps[i] * ei_w[d];
    }
#pragma unroll
    for (int o = 16; o >= 1; o >>= 1) dot += __shfl_xor(dot, o, 32);
    float g = sigmoid_f(dot + ei_b[0]);

    float u[3] = { unit[e * 3 + 0], unit[e * 3 + 1], unit[e * 3 + 2] };
#pragma unroll
    for (int i = 0; i < 4; ++i) {
        int d = lane + 32 * i;
        atomicAdd(&s2[(size_t)dst * DIM + d], ps[i] * g);
#pragma unroll
        for (int c = 0; c < 3; ++c) {
            float vns = vn[((size_t)src * 3 + c) * DIM + d];
            float m = (vns * pvv[i] + pvs[i] * u[c]) * g;
            atomicAdd(&v2[((size_t)dst * 3 + c) * DIM + d], m);
        }
    }
}

// ---------------- || Vv ||_c  and concat [Vv_norm, s]
__global__ void vnorm_cat_kernel(const float* __restrict__ UVv, const float* __restrict__ s2,
                                 float* __restrict__ cat)
{
    int i = blockIdx.x * blockDim.x + threadIdx.x;
    if (i >= NN * DIM) return;
    int n = i / DIM, d = i % DIM;
    float acc = 1e-6f;
#pragma unroll
    for (int c = 0; c < 3; ++c) {
        float x = UVv[((size_t)n * 3 + c) * D2 + DIM + d];
        acc += x * x;
    }
    cat[(size_t)n * D2 + d] = sqrtf(acc);
    cat[(size_t)n * D2 + DIM + d] = s2[i];
}

// ---------------- final update: inner products, gated residuals
__global__ void final_update_kernel(const float* __restrict__ UVv, const float* __restrict__ ab,
                                    const float* __restrict__ s2, const float* __restrict__ v2,
                                    float* __restrict__ s_out, float* __restrict__ v_out)
{
    int i = blockIdx.x * blockDim.x + threadIdx.x;
    if (i >= NN * DIM) return;
    int n = i / DIM, d = i % DIM;
    float avv = ab[(size_t)n * D3 + d];
    float asv = ab[(size_t)n * D3 + DIM + d];
    float ass = ab[(size_t)n * D3 + 2 * DIM + d];
    float inner = 0.0f;
#pragma unroll
    for (int c = 0; c < 3; ++c) {
        size_t r = (size_t)n * 3 + c;
        float Uv = UVv[r * D2 + d];
        float Vv = UVv[r * D2 + DIM + d];
        inner += Uv * Vv;
        v_out[r * DIM + d] = v2[r * DIM + d] + avv * Uv;
    }
    s_out[i] = s2[i] + ass + asv * inner;
}

// =====================================================================
static inline void gemm(const float* A, const float* B, const float* bias, float* C,
                        int M, int K, int Nc, int epi, hipStream_t stream)
{
    dim3 grid(Nc / 64, (M + 127) / 128);
    if (epi) wmma_gemm_f32<1><<<grid, 256, 0, stream>>>(A, B, bias, C, M, K, Nc);
    else     wmma_gemm_f32<0><<<grid, 256, 0, stream>>>(A, B, bias, C, M, K, Nc);
}

extern "C" void kernel_launch(void* const* d_in, const int* in_sizes, int n_in,
                              void* d_out, int out_size, void* d_ws, size_t ws_size,
                              hipStream_t stream)
{
    // inputs (setup_inputs dict order, params flattened in insertion order)
    const float* t    = (const float*)d_in[0];
    const float* h    = (const float*)d_in[1];
    const float* pos  = (const float*)d_in[2];
    const int* nidx   = (const int*)d_in[3];
    const int* eidx   = (const int*)d_in[4];
    const float* fourier_W = (const float*)d_in[5];
    const float* node_emb  = (const float*)d_in[6];
    const float* ntp_w     = (const float*)d_in[7];
    const float* ntp_b     = (const float*)d_in[8];
    const float* W_w    = (const float*)d_in[9];
    const float* W_b    = (const float*)d_in[10];
    const float* msg1_w = (const float*)d_in[11];
    const float* msg1_b = (const float*)d_in[12];
    const float* msg2_w = (const float*)d_in[13];
    const float* msg2_b = (const float*)d_in[14];
    const float* ei_w   = (const float*)d_in[15];
    const float* ei_b   = (const float*)d_in[16];
    const float* ln_g   = (const float*)d_in[17];
    const float* ln_b   = (const float*)d_in[18];
    const float* ln_gv  = (const float*)d_in[19];
    const float* uv_w   = (const float*)d_in[20];
    const float* upd1_w = (const float*)d_in[21];
    const float* upd1_b = (const float*)d_in[22];
    const float* upd2_w = (const float*)d_in[23];
    const float* upd2_b = (const float*)d_in[24];

    // workspace layout (floats)
    float* ws = (float*)d_ws;
    size_t o = 0;
    float* s_st  = ws + o; o += (size_t)NN * DIM;
    float* v_st  = ws + o; o += (size_t)NN * 3 * DIM;
    float* sn    = ws + o; o += (size_t)NN * DIM;
    float* vn    = ws + o; o += (size_t)NN * 3 * DIM;
    float* s2    = ws + o; o += (size_t)NN * DIM;
    float* v2    = ws + o; o += (size_t)NN * 3 * DIM;
    float* h1    = ws + o; o += (size_t)NN * DIM;
    float* phi   = ws + o; o += (size_t)NN * D3;
    float* estat = ws + o; o += (size_t)NE * NRBF;
    float* unit  = ws + o; o += (size_t)NE * 3;
    float* Wbuf  = ws + o; o += (size_t)ECHUNK * D3;
    float* temb  = ws + o; o += (size_t)NG * DIM;
    float* tproj = ws + o; o += (size_t)NG * DIM;
    float* s0    = ws + o; o += (size_t)NN * DIM;
    float* UVv   = ws + o; o += (size_t)NN * 3 * D2;
    float* cat   = ws + o; o += (size_t)NN * D2;
    float* a1    = ws + o; o += (size_t)NN * DIM;
    float* abuf  = ws + o; o += (size_t)NN * D3;

    const int ND   = NN * DIM;
    const int ND3v = NN * 3 * DIM;

    // --- setup ---
    temb_kernel<<<(NG * DIM + 255) / 256, 256, 0, stream>>>(t, fourier_W, temb);
    // tproj = t_emb @ ntp_w[128:256,:] + ntp_b   [G,128]
    gemm(temb, ntp_w + 128 * DIM, ntp_b, tproj, NG, DIM, DIM, 0, stream);
    // s0 = h @ node_emb   [N,128], K=100
    gemm(h, node_emb, nullptr, s0, NN, HIN, DIM, 0, stream);
    // s = s0 @ ntp_w[0:128,:]
    gemm(s0, ntp_w, nullptr, s_st, NN, DIM, DIM, 0, stream);
    add_tproj_kernel<<<(ND + 255) / 256, 256, 0, stream>>>(s_st, tproj, nidx);
    zero_kernel<<<(ND3v + 255) / 256, 256, 0, stream>>>(v_st, ND3v);
    edge_embed_kernel<<<(NE + 255) / 256, 256, 0, stream>>>(pos, eidx, estat, unit);

    // --- layers ---
    for (int i = 0; i < NL; ++i) {
        norm_kernel<<<(NN + 7) / 8, 256, 0, stream>>>(
            s_st, v_st, ln_g + i * DIM, ln_b + i * DIM, ln_gv + i * DIM, sn, vn, s2, v2);

        // phi = silu(sn @ msg1 + b) @ msg2 + b
        gemm(sn, msg1_w + (size_t)i * DIM * DIM, msg1_b + i * DIM, h1, NN, DIM, DIM, 1, stream);
        gemm(h1, msg2_w + (size_t)i * DIM * D3, msg2_b + i * D3, phi, NN, DIM, D3, 0, stream);

        // edge GEMM + interact, chunked
        for (int e0 = 0; e0 < NE; e0 += ECHUNK) {
            gemm(estat + (size_t)e0 * NRBF, W_w + (size_t)i * NRBF * D3, W_b + i * D3,
                 Wbuf, ECHUNK, NRBF, D3, 0, stream);
            edge_interact_kernel<<<(ECHUNK + 7) / 8, 256, 0, stream>>>(
                Wbuf, phi, vn, unit, eidx, ei_w + i * DIM, ei_b + i, s2, v2, e0, ECHUNK);
        }

        // update layer
        gemm(v2, uv_w + (size_t)i * DIM * D2, nullptr, UVv, NN * 3, DIM, D2, 0, stream);
        vnorm_cat_kernel<<<(ND + 255) / 256, 256, 0, stream>>>(UVv, s2, cat);
        gemm(cat, upd1_w + (size_t)i * D2 * DIM, upd1_b + i * DIM, a1, NN, D2, DIM, 1, stream);
        gemm(a1, upd2_w + (size_t)i * DIM * D3, upd2_b + i * D3, abuf, NN, DIM, D3, 0, stream);

        float* so = (i == NL - 1) ? (float*)d_out : s_st;
        float* vo = (i == NL - 1) ? ((float*)d_out + (size_t)NN * DIM) : v_st;
        final_update_kernel<<<(ND + 255) / 256, 256, 0, stream>>>(UVv, abuf, s2, v2, so, vo);
    }
}